// VQEmbeddingGSSoft_73486890434985
// MI455X (gfx1250) — compile-verified
//
#include <hip/hip_runtime.h>
#include <math.h>

typedef __attribute__((ext_vector_type(2))) float v2f;
typedef __attribute__((ext_vector_type(8))) float v8f;

#define B_   32
#define C_   64
#define T_   4096
#define M_   512
#define D_   64
#define ROWS (B_*T_)                       // 131072 token rows
#define RPW  16                            // rows per wave
#define WAVES 4
#define RPB  (RPW*WAVES)                   // 64 rows per block
#define LDS_STRIDE 516                     // padded row stride (floats) -> no bank conflicts
#define TILE_FLOATS (RPW*LDS_STRIDE)       // 8256 floats per wave tile
#define LN_M 6.2383246250395075f           // ln(512)
#define INV_TEMP 2.0f                      // 1/0.5

__device__ __forceinline__ float wave_max(float v) {
#pragma unroll
  for (int off = 16; off > 0; off >>= 1) v = fmaxf(v, __shfl_xor(v, off, 32));
  return v;
}
__device__ __forceinline__ float wave_sum(float v) {
#pragma unroll
  for (int off = 16; off > 0; off >>= 1) v += __shfl_xor(v, off, 32);
  return v;
}

// ws layout: [0..511] = |w_m|^2, [512..1023] = avg_probs accumulator, [1024] = KL accumulator
__global__ void vq_init(const float* __restrict__ w, float* __restrict__ ws) {
  int m = threadIdx.x;                     // 512 threads
  const float* wr = w + m * D_;
  float s = 0.f;
#pragma unroll
  for (int d = 0; d < D_; ++d) s += wr[d] * wr[d];
  ws[m] = s;
  ws[M_ + m] = 0.f;
  if (m == 0) ws[2 * M_] = 0.f;
}

__global__ void __launch_bounds__(WAVES * 32)
vq_main(const float* __restrict__ x, const float* __restrict__ u,
        const float* __restrict__ w, float* __restrict__ ws,
        float* __restrict__ out) {
  extern __shared__ float smem[];
  float* bavg = smem + WAVES * TILE_FLOATS;   // 512 block-partial avg_probs
  float* bkl  = bavg + M_;                    // 1 block-partial KL

  const int tid  = threadIdx.x;
  const int wave = tid >> 5;
  const int lane = tid & 31;
  float* S = smem + wave * TILE_FLOATS;       // this wave's 16x512 tile (stride 516)

  for (int m = tid; m < M_; m += WAVES * 32) bavg[m] = 0.f;
  if (tid == 0) *bkl = 0.f;
  __syncthreads();

  const int i0w = blockIdx.x * RPB + wave * RPW;   // first global row of this wave
  const int b   = i0w >> 12;                        // / T_
  const int t0  = i0w & (T_ - 1);
  const int rlo = lane & 15;
  const int hi2 = (lane >> 4) * 2;                  // K sub-offset per WMMA f32 A layout

  // ---------------- GEMM1: dot = x_row . w_m  (16 x 512, K = 64) -------------
  // x is (B, C, T): column c of the A tile is 16 contiguous floats at x[b,c,t0..]
  v2f a[16];
  {
    const float* xb = x + (size_t)b * (size_t)(C_ * T_) + t0 + rlo;
#pragma unroll
    for (int ks = 0; ks < 16; ++ks) {
      int c = ks * 4 + hi2;
      a[ks].x = xb[(size_t)c * T_];
      a[ks].y = xb[(size_t)(c + 1) * T_];
    }
  }

  for (int nt = 0; nt < 32; ++nt) {                 // 32 N-tiles of 16 codes
    const int n = nt * 16 + rlo;
    const float* wr = w + n * D_;                   // B[k][n] = weight[n][k]
    v8f acc = {};
#pragma unroll
    for (int ks = 0; ks < 16; ++ks) {
      int c = ks * 4 + hi2;
      v2f bf; bf.x = wr[c]; bf.y = wr[c + 1];
      acc = __builtin_amdgcn_wmma_f32_16x16x4_f32(false, a[ks], false, bf,
                                                  (short)0, acc, false, false);
    }
    const float wsqv = ws[n];                       // |w_n|^2
#pragma unroll
    for (int j = 0; j < 8; ++j) {                   // logit = 2*dot - |w|^2 (row term cancels)
      int row = j + 8 * (lane >> 4);
      S[row * LDS_STRIDE + nt * 16 + rlo] = 2.f * acc[j] - wsqv;
    }
  }

  // -------- per-row: probs softmax (KL) + gumbel softmax (samples) -----------
  float avg_acc[16];
#pragma unroll
  for (int j = 0; j < 16; ++j) avg_acc[j] = 0.f;
  float kl_lane = 0.f;

  for (int r = 0; r < 16; ++r) {
    const size_t urow = (size_t)(i0w + r) * M_;     // u row is contiguous, read once
    float l[16], z[16];
    float ml = -3.4e38f, mz = -3.4e38f;
#pragma unroll
    for (int j = 0; j < 16; ++j) {
      int m = lane + 32 * j;
      float lv = S[r * LDS_STRIDE + m];
      float uu = u[urow + m];
      uu = fminf(fmaxf(uu, 1e-9f), 1.f - 1e-9f);
      float g  = -__logf(-__logf(uu));
      float zv = (lv + g) * INV_TEMP;
      l[j] = lv; z[j] = zv;
      ml = fmaxf(ml, lv); mz = fmaxf(mz, zv);
    }
    ml = wave_max(ml); mz = wave_max(mz);
    float sp = 0.f, sz = 0.f;
#pragma unroll
    for (int j = 0; j < 16; ++j) { sp += __expf(l[j] - ml); sz += __expf(z[j] - mz); }
    sp = wave_sum(sp); sz = wave_sum(sz);
    const float isp = 1.f / sp, isz = 1.f / sz, lsp = __logf(sp);
#pragma unroll
    for (int j = 0; j < 16; ++j) {
      float lm = l[j] - ml;
      float p  = __expf(lm) * isp;
      kl_lane += p * ((lm - lsp) + LN_M);
      float s  = __expf(z[j] - mz) * isz;
      S[r * LDS_STRIDE + lane + 32 * j] = s;        // samples overwrite logits in LDS
      avg_acc[j] += s;
    }
  }

  // ------------- GEMM2: quantized(16x64) = samples(16x512) @ w ---------------
  v8f q[4];
#pragma unroll
  for (int nt = 0; nt < 4; ++nt) q[nt] = (v8f){};
  for (int ks = 0; ks < 128; ++ks) {
    const int k0 = ks * 4 + hi2;
    v2f af; af.x = S[rlo * LDS_STRIDE + k0];        // padded stride -> conflict-free
    af.y = S[rlo * LDS_STRIDE + k0 + 1];
#pragma unroll
    for (int nt = 0; nt < 4; ++nt) {
      const int n = nt * 16 + rlo;
      v2f bf; bf.x = w[(size_t)k0 * D_ + n]; bf.y = w[(size_t)(k0 + 1) * D_ + n];
      q[nt] = __builtin_amdgcn_wmma_f32_16x16x4_f32(false, af, false, bf,
                                                    (short)0, q[nt], false, false);
    }
  }
#pragma unroll
  for (int nt = 0; nt < 4; ++nt)
#pragma unroll
    for (int j = 0; j < 8; ++j) {
      int row = j + 8 * (lane >> 4);
      out[(size_t)(i0w + row) * D_ + nt * 16 + rlo] = q[nt][j];
    }

  // ------------- block-level accumulation, then one global add per block -----
#pragma unroll
  for (int j = 0; j < 16; ++j) atomicAdd(&bavg[lane + 32 * j], avg_acc[j]);
  kl_lane = wave_sum(kl_lane);
  if (lane == 0) atomicAdd(bkl, kl_lane);
  __syncthreads();
  for (int m = tid; m < M_; m += WAVES * 32) atomicAdd(&ws[M_ + m], bavg[m]);
  if (tid == 0) atomicAdd(&ws[2 * M_], *bkl);
}

__global__ void vq_fin(const float* __restrict__ ws, float* __restrict__ out) {
  __shared__ float red[M_];
  int t = threadIdx.x;                              // 512 threads
  float ap = ws[M_ + t] * (1.f / (float)ROWS);      // avg_probs
  red[t] = ap * __logf(ap + 1e-10f);
  __syncthreads();
  for (int s = M_ / 2; s > 0; s >>= 1) {
    if (t < s) red[t] += red[t + s];
    __syncthreads();
  }
  if (t == 0) {
    out[(size_t)ROWS * D_]     = ws[2 * M_] * (1.f / (float)B_);  // KL
    out[(size_t)ROWS * D_ + 1] = __expf(-red[0]);                 // perplexity
  }
}

extern "C" void kernel_launch(void* const* d_in, const int* in_sizes, int n_in,
                              void* d_out, int out_size, void* d_ws, size_t ws_size,
                              hipStream_t stream) {
  const float* x = (const float*)d_in[0];   // (32, 64, 4096)
  const float* u = (const float*)d_in[1];   // (32, 4096, 512)
  const float* w = (const float*)d_in[2];   // (512, 64)
  float* out = (float*)d_out;               // 131072*64 quantized + KL + perplexity
  float* ws  = (float*)d_ws;                // >= (2*512+1)*4 bytes used

  const size_t smem = (size_t)(WAVES * TILE_FLOATS + M_ + 16) * sizeof(float); // ~134 KB
  hipFuncSetAttribute(reinterpret_cast<const void*>(vq_main),
                      hipFuncAttributeMaxDynamicSharedMemorySize, (int)smem);

  vq_init<<<1, M_, 0, stream>>>(w, ws);
  vq_main<<<ROWS / RPB, WAVES * 32, smem, stream>>>(x, u, w, ws, out);
  vq_fin<<<1, M_, 0, stream>>>(ws, out);
}